// GNNLayer_7361573945709
// MI455X (gfx1250) — compile-verified
//
#include <hip/hip_runtime.h>

#define N_NODES 100000
#define N_EDGES 1600000
#define D 128

typedef __attribute__((ext_vector_type(2))) float v2f;
typedef __attribute__((ext_vector_type(8))) float v8f;

// Guaranteed-native f32 atomic add (no CAS-loop expansion): fire-and-forget,
// tracked by STOREcnt; s_endpgm's implicit wait-idle covers completion.
__device__ __forceinline__ void atomic_add_f32_native(float* p, float v) {
    asm volatile("global_atomic_add_f32 %0, %1, off"
                 :
                 : "v"(p), "v"(v)
                 : "memory");
}

// ---------------------------------------------------------------------------
// Phase 1: edge scatter.  One wave32 per edge; lane l owns floats [4l, 4l+4).
// agg[dst, :] += w_e * x[src, :]
// x (51.2 MB) and agg (51.2 MB) are both L2-resident (192 MB L2), so the
// ~819 MB of gathers and ~819 MB of atomic traffic stay on-die.
// ---------------------------------------------------------------------------
__global__ void gnn_scatter_kernel(const float* __restrict__ x,
                                   const long long* __restrict__ edge_index,
                                   const float* __restrict__ edge_weight,
                                   float* __restrict__ agg) {
    const int lane = threadIdx.x & 31;
    const int waveInBlock = threadIdx.x >> 5;
    const long long e = (long long)blockIdx.x * (blockDim.x >> 5) + waveInBlock;
    if (e >= N_EDGES) return;

    const long long src = edge_index[e];             // row 0
    const long long dst = edge_index[N_EDGES + e];   // row 1
    const float w = edge_weight[e];

    const float4 v = ((const float4*)(x + src * D))[lane];   // global_load_b128 gather
    float* ap = agg + dst * (long long)D + lane * 4;

    atomic_add_f32_native(ap + 0, v.x * w);
    atomic_add_f32_native(ap + 1, v.y * w);
    atomic_add_f32_native(ap + 2, v.z * w);
    atomic_add_f32_native(ap + 3, v.w * w);
}

// ---------------------------------------------------------------------------
// Phase 2: out = agg @ W^T + b using V_WMMA_F32_16X16X4_F32.
// Block = 256 threads = 8 waves.  blockIdx.x selects a 16-row node strip;
// wave w computes output columns [16w, 16w+16).  K loop: 32 x (k=4) WMMAs.
//
// A fragment (16x4 f32, 2 VGPRs/lane):
//   lane<16 : A[M=lane][K=k0+0], A[M=lane][K=k0+1]
//   lane>=16: A[M=lane-16][K=k0+2], A[M=lane-16][K=k0+3]
// B fragment (4x16 f32 = W^T tile): B[k][n] = W[n][k], same contiguous pattern.
// C/D: VGPR r -> row m0 + r + 8*(lane/16), col n0 + (lane&15).
// ---------------------------------------------------------------------------
__global__ void gnn_gemm_wmma_kernel(const float* __restrict__ agg,
                                     const float* __restrict__ Wm,
                                     const float* __restrict__ bias,
                                     float* __restrict__ out) {
    const int lane = threadIdx.x & 31;
    const int wave = threadIdx.x >> 5;         // 0..7
    const int m0 = blockIdx.x * 16;            // node row tile (100000 % 16 == 0)
    const int n0 = wave * 16;                  // output feature tile
    const int lrow = lane & 15;
    const int khalf = (lane >> 4) * 2;         // 0 or 2

    // Initialize accumulator with bias (constant per output column).
    const float bn = bias[n0 + lrow];
    v8f c;
#pragma unroll
    for (int i = 0; i < 8; ++i) c[i] = bn;

    const float* aBase = agg + (long long)(m0 + lrow) * D + khalf;
    const float* bBase = Wm + (long long)(n0 + lrow) * D + khalf;

#pragma unroll 8
    for (int k0 = 0; k0 < D; k0 += 4) {
        v2f a = *(const v2f*)(aBase + k0);     // global_load_b64
        v2f b = *(const v2f*)(bBase + k0);     // global_load_b64 (L2-hot, 64KB)
        c = __builtin_amdgcn_wmma_f32_16x16x4_f32(
                /*neg_a=*/false, a, /*neg_b=*/false, b,
                /*c_mod=*/(short)0, c, /*reuse_a=*/false, /*reuse_b=*/false);
    }

    float* oBase = out + (long long)(m0 + (lane >> 4) * 8) * D + n0 + lrow;
#pragma unroll
    for (int r = 0; r < 8; ++r) oBase[(long long)r * D] = c[r];
}

// ---------------------------------------------------------------------------
// Inputs (setup_inputs order): x[100000*128] f32, edge_index[2*1600000] i64,
// edge_weight[1600000] f32, W[128*128] f32, b[128] f32, vertex_cnt, rule_cnt.
// d_out: 100000*128 f32.  d_ws: agg accumulator (51.2 MB).
// ---------------------------------------------------------------------------
extern "C" void kernel_launch(void* const* d_in, const int* in_sizes, int n_in,
                              void* d_out, int out_size, void* d_ws, size_t ws_size,
                              hipStream_t stream) {
    const float*     x  = (const float*)d_in[0];
    const long long* ei = (const long long*)d_in[1];
    const float*     ew = (const float*)d_in[2];
    const float*     Wm = (const float*)d_in[3];
    const float*     b  = (const float*)d_in[4];
    float* out = (float*)d_out;
    float* agg = (float*)d_ws;

    // Zero the aggregation buffer (capture-safe, deterministic every call).
    hipMemsetAsync(agg, 0, (size_t)N_NODES * D * sizeof(float), stream);

    // Scatter: 8 edges (waves) per 256-thread block.
    const int wavesPerBlock = 8;
    const int scatterBlocks = (N_EDGES + wavesPerBlock - 1) / wavesPerBlock;  // 200000
    gnn_scatter_kernel<<<scatterBlocks, 256, 0, stream>>>(x, ei, ew, agg);

    // GEMM: one block per 16-node strip, 8 waves cover all 128 output features.
    gnn_gemm_wmma_kernel<<<N_NODES / 16, 256, 0, stream>>>(agg, Wm, b, out);
}